// NaiveBiasBase_14688788152578
// MI455X (gfx1250) — compile-verified
//
#include <hip/hip_runtime.h>
#include <hip/hip_bf16.h>

// Toeplitz bias construction for MI455X (gfx1250).
//
// out[b,h,i,j] = exp(w_[h, (L-1)-i+j] - offset[h]),  L=2048, H=16, batch=2.
// Pure store-bandwidth problem: 512 MiB out / 23.3 TB/s ~= 23 us floor.
//   K1: wexp[h,k] = exp(w_[h,k]-offset[h]) into d_ws (262 KB, L2-resident).
//   K2: per (h,i) block: async-copy the 8KB window wexp[h, L-1-i .. 2L-2-i]
//       into LDS (global_load_async_to_lds_b32 -> ASYNCcnt -> s_wait_asynccnt),
//       then stream to both batches with ds_load_b128 + 128-bit NT stores.

typedef float f32x4 __attribute__((ext_vector_type(4)));

#define TPB 256

#if __has_builtin(__builtin_amdgcn_global_load_async_to_lds_b32)
#define HAVE_ASYNC_LDS 1
#else
#define HAVE_ASYNC_LDS 0
#endif

typedef __attribute__((address_space(1))) int g_i32;
typedef __attribute__((address_space(3))) int l_i32;

__device__ __forceinline__ void wait_async_zero() {
#if __has_builtin(__builtin_amdgcn_s_wait_asynccnt)
    __builtin_amdgcn_s_wait_asynccnt(0);
#else
    asm volatile("s_wait_asynccnt 0" ::: "memory");
#endif
}

__device__ __forceinline__ void stage_one_to_lds(const float* gsrc, float* ldst) {
#if HAVE_ASYNC_LDS
    __builtin_amdgcn_global_load_async_to_lds_b32(
        (g_i32*)(unsigned long long)gsrc,
        (l_i32*)(unsigned)(unsigned long long)ldst,
        0, 0);
#else
    *ldst = *gsrc;
#endif
}

// ---- K2 (specialized, shown first so its disasm is visible) --------------
// grid = (L rows, H heads), block = TPB, static 8KB LDS.
template <int L, int BATCH>
__global__ void bias_fill_tpl(const float* __restrict__ wexp,
                              float* __restrict__ out,
                              int twoLm1, long long perBatch) {
    __shared__ float lds[L];
    const int i = blockIdx.x;   // row
    const int h = blockIdx.y;   // head
    const int t = threadIdx.x;

    // Stage wexp[h, (L-1-i) .. (L-1-i)+L-1] -> lds[0..L-1]  (4B-aligned only).
    const long long start = (long long)h * twoLm1 + (L - 1 - i);
#pragma unroll
    for (int s = 0; s < L; s += TPB)
        stage_one_to_lds(wexp + start + s + t, &lds[s + t]);
#if HAVE_ASYNC_LDS
    wait_async_zero();
#endif
    __syncthreads();

    // 16B-aligned LDS reads, 128-bit non-temporal global stores.
    const f32x4* lds4 = (const f32x4*)lds;
    const long long rowbase = ((long long)h * L + i) * (long long)L;
#pragma unroll
    for (int jj = 0; jj < L / 4; jj += TPB) {
        const int j4 = jj + t;
        f32x4 v = lds4[j4];
#pragma unroll
        for (int b = 0; b < BATCH; ++b) {
            f32x4* dst = (f32x4*)(out + (long long)b * perBatch + rowbase + 4LL * j4);
            __builtin_nontemporal_store(v, dst);
        }
    }
}

// ---- K1: tiny exp table (2-D grid: y = head, no integer division) --------
__global__ void bias_exp_kernel(const float* __restrict__ w_,
                                const float* __restrict__ offset,
                                float* __restrict__ wexp,
                                int twoLm1) {
    const int h = blockIdx.y;
    const int k = blockIdx.x * blockDim.x + threadIdx.x;
    if (k < twoLm1) {
        const long long p = (long long)h * twoLm1 + k;
        wexp[p] = expf(w_[p] - offset[h]);
    }
}

// ---- K2 (generic shapes) -------------------------------------------------
__global__ void bias_fill_gen(const float* __restrict__ wexp,
                              float* __restrict__ out,
                              int L, int twoLm1, int batch,
                              long long perBatch) {
    extern __shared__ float lds[];
    const int i = blockIdx.x;
    const int h = blockIdx.y;
    const int t = threadIdx.x;

    const long long start = (long long)h * twoLm1 + (L - 1 - i);
    for (int s = t; s < L; s += TPB)
        stage_one_to_lds(wexp + start + s, &lds[s]);
#if HAVE_ASYNC_LDS
    wait_async_zero();
#endif
    __syncthreads();

    const f32x4* lds4 = (const f32x4*)lds;
    const long long rowbase = ((long long)h * L + i) * (long long)L;
    for (int j4 = t; j4 < (L >> 2); j4 += TPB) {
        f32x4 v = lds4[j4];
        for (int b = 0; b < batch; ++b) {
            f32x4* dst = (f32x4*)(out + (long long)b * perBatch + rowbase + 4LL * j4);
            __builtin_nontemporal_store(v, dst);
        }
    }
}

// ---- Fallback: fused naive (if d_ws is too small or L is unsupported) -------------------
__global__ void bias_naive_kernel(const float* __restrict__ w_,
                                  const float* __restrict__ offset,
                                  float* __restrict__ out,
                                  int L, int twoLm1, long long perBatch,
                                  long long total) {
    long long stride = (long long)gridDim.x * blockDim.x;
    for (long long idx = (long long)blockIdx.x * blockDim.x + threadIdx.x;
         idx < total; idx += stride) {
        long long rem = idx % perBatch;
        int h = (int)(rem / ((long long)L * L));
        long long r2 = rem % ((long long)L * L);
        int i = (int)(r2 / L);
        int j = (int)(r2 % L);
        out[idx] = expf(w_[(long long)h * twoLm1 + (L - 1 - i + j)] - offset[h]);
    }
}

extern "C" void kernel_launch(void* const* d_in, const int* in_sizes, int n_in,
                              void* d_out, int out_size, void* d_ws, size_t ws_size,
                              hipStream_t stream) {
    const float* w_     = (const float*)d_in[0];
    const float* offset = (const float*)d_in[1];
    float* out = (float*)d_out;

    const int H      = in_sizes[1];             // 16
    const int twoLm1 = in_sizes[0] / H;         // 4095
    const int L      = (twoLm1 + 1) / 2;        // 2048
    const long long perBatch = (long long)H * L * L;
    const int batch = (int)((long long)out_size / perBatch);  // 2

    const size_t need = (size_t)H * twoLm1 * sizeof(float);
    if (ws_size >= need && L >= 4 && (L % (4 * TPB)) == 0 && L <= 65536) {
        float* wexp = (float*)d_ws;
        dim3 egrid((twoLm1 + TPB - 1) / TPB, H);
        bias_exp_kernel<<<egrid, TPB, 0, stream>>>(w_, offset, wexp, twoLm1);

        dim3 grid(L, H);
        if (L == 2048 && batch == 2) {
            bias_fill_tpl<2048, 2><<<grid, TPB, 0, stream>>>(
                wexp, out, twoLm1, perBatch);
        } else {
            bias_fill_gen<<<grid, TPB, L * sizeof(float), stream>>>(
                wexp, out, L, twoLm1, batch, perBatch);
        }
    } else {
        const long long total = (long long)batch * perBatch;
        long long blocks = (total + TPB - 1) / TPB;
        if (blocks > 1048576LL) blocks = 1048576LL;
        bias_naive_kernel<<<(int)blocks, TPB, 0, stream>>>(
            w_, offset, out, L, twoLm1, perBatch, total);
    }
}